// CTCLoss_25795573580211
// MI455X (gfx1250) — compile-verified
//
#include <hip/hip_runtime.h>
#include <stdint.h>

#define TT 2048
#define NN 64
#define CC 128
#define SS 256
#define LL 513            // 2*S + 1 extended states
#define NEGF (-1e30f)

// One workgroup per batch sample; 288 threads = 9 wave32s.
// Thread i owns extended states (2i, 2i+1) in registers (i <= 255),
// thread 256 owns state 512 only. Only ODD alpha values are shared via LDS
// (blank states never serve as a 2-back source). log_prob slabs (128 floats
// per (t,n)) streamed via CDNA5 async-to-LDS, depth-2 pipeline, ASYNCcnt.
__global__ __launch_bounds__(288) void ctc_alpha_kernel(
    const float* __restrict__ log_probs,     // (T, N, C)
    const int*   __restrict__ targets,       // (N, S)
    const int*   __restrict__ input_lengths, // (N,)
    const int*   __restrict__ target_lengths,// (N,)
    float*       __restrict__ per_sample)    // (N,) loss / target_len
{
    __shared__ float aOdd[2][SS + 8];        // odd-state alpha, double buffered
    __shared__ float lp[4][CC];              // 4 rotating log_prob slabs
    __shared__ float afin[LL + 7];           // final alpha dump

    const int n   = blockIdx.x;
    const int tid = threadIdx.x;

    const int len = input_lengths[n];
    const int tl  = target_lengths[n];

    const bool hasEven = (tid <= 256);       // state 2*tid   (<= 512)
    const bool hasOdd  = (tid <= 255);       // state 2*tid+1 (<= 511)

    // Odd-state label and 2-back legality (even states are BLANK: never can2).
    int  e    = 0;
    bool can2 = false;
    if (hasOdd) {
        e = targets[n * SS + tid];
        if (tid >= 1) {
            int em2 = targets[n * SS + tid - 1];
            can2 = (e != em2);
        }
    }

    const float* gbase = log_probs + (size_t)n * CC;

    // --- prologue: async-prefetch slabs 0 and 1 (wave 0: 32 lanes x 16B) ---
    if (tid < 32) {
        {
            unsigned     l0 = (unsigned)(uintptr_t)&lp[0][tid * 4];
            const float* g0 = gbase + (size_t)0 * NN * CC + tid * 4;
            asm volatile("global_load_async_to_lds_b128 %0, %1, off"
                         :: "v"(l0), "v"(g0) : "memory");
        }
        {
            unsigned     l1 = (unsigned)(uintptr_t)&lp[1][tid * 4];
            const float* g1 = gbase + (size_t)1 * NN * CC + tid * 4;
            asm volatile("global_load_async_to_lds_b128 %0, %1, off"
                         :: "v"(l1), "v"(g1) : "memory");
        }
        asm volatile("s_wait_asynccnt 0x1" ::: "memory"); // slab 0 resident
    }
    __syncthreads();

    // --- t = 0 init ---
    float e0 = NEGF;                          // alpha[2*tid]
    float o1 = NEGF;                          // alpha[2*tid+1]
    if (tid == 0) {
        e0 = lp[0][0];                        // state 0 = blank
        o1 = (tl > 0) ? lp[0][e] : NEGF;      // state 1 = first label
    }
    if (hasOdd) aOdd[0][tid] = o1;

    // --- main recursion: one barrier per timestep ---
    for (int t = 1; t < TT; ++t) {
        if (tid < 32) {
            if (t + 1 < TT) {
                unsigned     lo = (unsigned)(uintptr_t)&lp[(t + 1) & 3][tid * 4];
                const float* gp = gbase + (size_t)(t + 1) * NN * CC + tid * 4;
                asm volatile("global_load_async_to_lds_b128 %0, %1, off"
                             :: "v"(lo), "v"(gp) : "memory");
                asm volatile("s_wait_asynccnt 0x1" ::: "memory"); // slab t ready
            } else {
                asm volatile("s_wait_asynccnt 0x0" ::: "memory");
            }
        }
        __syncthreads();  // publishes slab t and odd-alpha[t-1]; WAR fence

        if (hasEven) {
            const float* aprev = aOdd[(t - 1) & 1];
            const float  aO    = (tid >= 1) ? aprev[tid - 1] : NEGF; // alpha[2i-1]
            const float  lpB   = lp[t & 3][0];                       // blank emission
            const bool   live  = (t < len);

            // even state 2i: logaddexp(e0, aO) + lp_blank  (2-back illegal)
            {
                float m  = fmaxf(e0, aO);
                float sm = __expf(e0 - m) + __expf(aO - m);
                float v  = m + __logf(sm) + lpB;
                float ne = live ? v : e0;
                // odd state 2i+1: logaddexp3(o1, e0_old, can2 ? aO : NEG) + lp[e]
                if (hasOdd) {
                    float a2 = can2 ? aO : NEGF;
                    float m1 = fmaxf(o1, fmaxf(e0, a2));
                    float s1 = __expf(o1 - m1) + __expf(e0 - m1) + __expf(a2 - m1);
                    float v1 = m1 + __logf(s1) + lp[t & 3][e];
                    o1 = live ? v1 : o1;
                    aOdd[t & 1][tid] = o1;
                }
                e0 = ne;
            }
        }
    }
    __syncthreads();

    // dump final alpha so thread 0 can read arbitrary indices
    if (hasEven) afin[2 * tid] = e0;
    if (hasOdd)  afin[2 * tid + 1] = o1;
    __syncthreads();

    if (tid == 0) {
        int   i1 = 2 * tl;                          // last blank
        int   i2 = (i1 - 1 > 0) ? (i1 - 1) : 0;     // last label
        float v1 = afin[i1];
        float v2 = (tl > 0) ? afin[i2] : NEGF;
        float m  = fmaxf(v1, v2);
        float loss = -(m + __logf(__expf(v1 - m) + __expf(v2 - m)));
        if (loss >= 1e29f) loss = 0.0f;             // zero_infinity
        float d = (tl > 0) ? (float)tl : 1.0f;
        per_sample[n] = loss / d;
    }
}

__global__ void ctc_reduce_kernel(const float* __restrict__ per_sample,
                                  float* __restrict__ out)
{
    if (threadIdx.x == 0) {
        float acc = 0.0f;
        for (int i = 0; i < NN; ++i) acc += per_sample[i];
        out[0] = acc / (float)NN;
    }
}

extern "C" void kernel_launch(void* const* d_in, const int* in_sizes, int n_in,
                              void* d_out, int out_size, void* d_ws, size_t ws_size,
                              hipStream_t stream) {
    const float* log_probs      = (const float*)d_in[0];
    const int*   targets        = (const int*)  d_in[1];
    const int*   input_lengths  = (const int*)  d_in[2];
    const int*   target_lengths = (const int*)  d_in[3];
    float*       out            = (float*)d_out;
    float*       ws             = (float*)d_ws;   // >= NN floats

    hipLaunchKernelGGL(ctc_alpha_kernel, dim3(NN), dim3(288), 0, stream,
                       log_probs, targets, input_lengths, target_lengths, ws);
    hipLaunchKernelGGL(ctc_reduce_kernel, dim3(1), dim3(32), 0, stream, ws, out);
}